// BLTSmoothnessLoss_83958020702369
// MI455X (gfx1250) — compile-verified
//
#include <hip/hip_runtime.h>
#include <hip/hip_bf16.h>
#include <stdint.h>

// Problem constants from the reference
#define FRAD 3
#define KSZ  7
#define BT   32                 // output tile edge per block (32x32, 2x2 per thread)
#define HT2  (BT + 2*FRAD)      // 38: tile + halo

// 50 * log2(e)  (inv_2sr2 = 1/(2*0.1^2) = 50, folded for exp2)
#define K_R   72.134752044f
// 0.5 * log2(e) for the spatial Gaussian exponent
#define K_S   0.72134752f
// log2( (sum_{x=-3..3} e^{-x^2/2})^2 ) = log2(6.2797848) : Gaussian normalization
#define LOG2S 2.6507353f
// per-tap exponent constant (k,l are compile-time after full unroll)
#define QV(k,l) (K_S * (float)(((k)-FRAD)*((k)-FRAD) + ((l)-FRAD)*((l)-FRAD)) + LOG2S)

typedef __attribute__((ext_vector_type(2))) float v2f;
typedef __attribute__((ext_vector_type(8))) float v8f;

__device__ __forceinline__ int reflect_idx(int i, int n) {
  i = (i < 0) ? -i : i;                 // jnp 'reflect' (no edge repeat)
  i = (i >= n) ? (2 * n - 2 - i) : i;
  return i;
}

// One bilateral tap: d = c - nb; w = exp2(-(50*log2e*ssd + q)); acc += |d|_1 * w
__device__ __forceinline__ void tap(const float4& c, const float4& nb, float q, float& acc) {
  const float d0 = c.x - nb.x;
  const float d1 = c.y - nb.y;
  const float d2 = c.z - nb.z;
  const float ssd = fmaf(d2, d2, fmaf(d1, d1, d0 * d0));
  const float w = __builtin_amdgcn_exp2f(-fmaf(ssd, K_R, q));
  acc = fmaf(fabsf(d0) + fabsf(d1) + fabsf(d2), w, acc);
}

__global__ __launch_bounds__(256) void blt_tile_kernel(
    const float* __restrict__ in, float* __restrict__ partials,
    int Hn, int Wn)
{
  __shared__ float4 tile[HT2][HT2];  // .x/.y/.z = channels, .w pad (16B align -> ds_load_b128)
  __shared__ float  red[256];

  const int tid = threadIdx.x;
  const int tx2 = (tid & 15) << 1;   // 2x2 register blocking
  const int ty2 = (tid >> 4) << 1;
  const int x0  = blockIdx.x * BT;
  const int y0  = blockIdx.y * BT;
  const long HW = (long)Hn * Wn;
  const float* __restrict__ base = in + (long)blockIdx.z * 3 * HW;

  // ---- Stage 38x38 halo tile (3 channels) into LDS via CDNA5 async copies ----
  // Reflect padding folded into per-lane global addresses; tracked by ASYNCcnt.
  for (int idx = tid; idx < HT2 * HT2; idx += 256) {
    const int ly = idx / HT2;
    const int lx = idx - ly * HT2;
    const int gy = reflect_idx(y0 + ly - FRAD, Hn);
    const int gx = reflect_idx(x0 + lx - FRAD, Wn);
    const float* p0 = base + (long)gy * Wn + gx;
    const unsigned l0 = (unsigned)(uintptr_t)&tile[ly][lx];  // low 32 bits = LDS byte offset
    const uint64_t g0 = (uint64_t)(uintptr_t)p0;
    const uint64_t g1 = (uint64_t)(uintptr_t)(p0 + HW);
    const uint64_t g2 = (uint64_t)(uintptr_t)(p0 + 2 * HW);
    asm volatile("global_load_async_to_lds_b32 %0, %1, off" :: "v"(l0),      "v"(g0) : "memory");
    asm volatile("global_load_async_to_lds_b32 %0, %1, off" :: "v"(l0 + 4u), "v"(g1) : "memory");
    asm volatile("global_load_async_to_lds_b32 %0, %1, off" :: "v"(l0 + 8u), "v"(g2) : "memory");
  }
#if defined(__has_builtin) && __has_builtin(__builtin_amdgcn_s_wait_asynccnt)
  __builtin_amdgcn_s_wait_asynccnt(0);
#else
  asm volatile("s_wait_asynccnt 0x0" ::: "memory");
#endif
  __syncthreads();

  // ---- 2x2-blocked 7x7 bilateral stencil ----
  // The four 7x7 neighborhoods share one 8x8 window: 64 ds_load_b128 per 4 pixels
  // (16/pixel vs 49/pixel unblocked) -> LDS is no longer the WGP bottleneck.
  const float4 c00 = tile[ty2 + FRAD][tx2 + FRAD];
  const float4 c01 = tile[ty2 + FRAD][tx2 + FRAD + 1];
  const float4 c10 = tile[ty2 + FRAD + 1][tx2 + FRAD];
  const float4 c11 = tile[ty2 + FRAD + 1][tx2 + FRAD + 1];
  float a00 = 0.f, a01 = 0.f, a10 = 0.f, a11 = 0.f;

#pragma unroll
  for (int k = 0; k < KSZ + 1; ++k) {
    float4 row[KSZ + 1];
#pragma unroll
    for (int l = 0; l < KSZ + 1; ++l) row[l] = tile[ty2 + k][tx2 + l];
#pragma unroll
    for (int l = 0; l < KSZ + 1; ++l) {
      // neighbor (k,l) feeds output (py,px) with tap (k-py, l-px) when in [0,6]
      if (k < KSZ  && l < KSZ ) tap(c00, row[l], QV(k,     l    ), a00);
      if (k < KSZ  && l >= 1  ) tap(c01, row[l], QV(k,     l - 1), a01);
      if (k >= 1   && l < KSZ ) tap(c10, row[l], QV(k - 1, l    ), a10);
      if (k >= 1   && l >= 1  ) tap(c11, row[l], QV(k - 1, l - 1), a11);
    }
  }
  const float acc = (a00 + a01) + (a10 + a11);

  // ---- per-block reduction (wave32-aware) ----
  red[tid] = acc;
  __syncthreads();
  if (tid < 128) red[tid] += red[tid + 128];
  __syncthreads();
  if (tid < 64)  red[tid] += red[tid + 64];
  __syncthreads();
  if (tid < 32) {
    float v = red[tid] + red[tid + 32];
#pragma unroll
    for (int off = 16; off > 0; off >>= 1) v += __shfl_down(v, off, 32);
    if (tid == 0) {
      partials[((long)blockIdx.z * gridDim.y + blockIdx.y) * gridDim.x + blockIdx.x] = v;
    }
  }
}

// Deterministic final reduction on one wave32; collapses the 32 lane sums with
// V_WMMA_F32_16X16X4_F32 (A = lane sums in K=0/K=2 columns, B = ones):
// D[m][n] = s_m + s_{m+16}; summing the 8 accumulator VGPRs and adding
// lane0 + lane16 yields the full 32-lane sum.
__global__ __launch_bounds__(32) void blt_reduce_kernel(
    const float* __restrict__ partials, int n, float* __restrict__ out, float scale)
{
  const int lane = threadIdx.x;
  float s = 0.f;
  for (int i = lane; i < n; i += 32) s += partials[i];   // n % 32 == 0: no divergence
#if defined(__has_builtin) && __has_builtin(__builtin_amdgcn_wmma_f32_16x16x4_f32)
  v2f a;  a.x = s;  a.y = 0.f;          // K=0 (lanes 0-15) / K=2 (lanes 16-31) columns
  v2f bm; bm.x = 1.f; bm.y = 1.f;       // ones: layout-independent
  v8f c = {};
  v8f d = __builtin_amdgcn_wmma_f32_16x16x4_f32(false, a, false, bm, (short)0, c, false, false);
  const float t = d[0] + d[1] + d[2] + d[3] + d[4] + d[5] + d[6] + d[7];
  const float total = __shfl(t, 0, 32) + __shfl(t, 16, 32);
#else
  float v = s;
#pragma unroll
  for (int off = 16; off > 0; off >>= 1) v += __shfl_down(v, off, 32);
  const float total = __shfl(v, 0, 32);
#endif
  if (lane == 0) out[0] = total * scale;
}

extern "C" void kernel_launch(void* const* d_in, const int* in_sizes, int n_in,
                              void* d_out, int out_size, void* d_ws, size_t ws_size,
                              hipStream_t stream) {
  const float* in  = (const float*)d_in[0];
  float* out       = (float*)d_out;
  float* partials  = (float*)d_ws;

  const int Bn = 8, Cn = 3, Hn = 512, Wn = 512;
  dim3 grid(Wn / BT, Hn / BT, Bn);          // 16 x 16 x 8 = 2048 tiles
  blt_tile_kernel<<<grid, 256, 0, stream>>>(in, partials, Hn, Wn);

  const int nblocks = (Wn / BT) * (Hn / BT) * Bn;
  const float scale = 1.0f / (float)((long)Bn * Cn * Hn * Wn);
  blt_reduce_kernel<<<1, 32, 0, stream>>>(partials, nblocks, out, scale);
}